// BatchMegaDecode_80427557585272
// MI455X (gfx1250) — compile-verified
//
#include <hip/hip_runtime.h>
#include <math.h>

// ---------------- model constants (fixed by the reference) ----------------
#define LAYERS 32
#define Hh     1024
#define H3     3072
#define NHEAD  16
#define HDIM   64
#define Ff     2048
#define SEQ    2048
#define BATCH  2
#define EPSV   1e-6f
#define NSPLIT 8           // K-split factor for GEMV occupancy

typedef __attribute__((ext_vector_type(2))) float v2f;
typedef __attribute__((ext_vector_type(8))) float v8f;

__device__ __forceinline__ float gelu_tanh(float x) {
    const float c = 0.7978845608028654f;  // sqrt(2/pi)
    float inner = c * (x + 0.044715f * x * x * x);
    return 0.5f * x * (1.0f + tanhf(inner));
}

// ---------------------------------------------------------------------------
// Split-K GEMV via fp32 WMMA (V_WMMA_F32_16X16X4_F32).
// Grid: x = row tiles (4 waves/block, 16 rows/wave), y = K chunk. No divisions.
// A-operand: 16x4 fp32 weight tile (lane l%16 = row, lane half -> K pair).
// B-operand: activation replicated into 16 N-columns; only N=0,1 real (B=2).
// Partials per chunk; deterministic reduce kernel finishes (no float atomics).
// ---------------------------------------------------------------------------
__global__ __launch_bounds__(128)
void gemv_wmma_partial(const float* __restrict__ W,
                       const float* __restrict__ act,     // [BATCH][K]
                       float* __restrict__ part,          // [NSPLIT][BATCH][rows]
                       int rows, int K) {
    int lane = threadIdx.x & 31;
    int lm   = lane & 15;        // A row-within-tile / B,C,D column N
    int half = lane >> 4;        // selects K pair {0,1} vs {2,3}

    int rowwave  = (blockIdx.x << 2) + (threadIdx.x >> 5);
    int chunk    = blockIdx.y;
    int row_base = rowwave << 4;
    if (row_base >= rows) return;

    int klen   = K >> 3;         // K / NSPLIT, multiple of 16 for all shapes here
    int kbegin = chunk * klen;
    int kend   = kbegin + klen;

    const float* wrow = W + (size_t)(row_base + lm) * (size_t)K + 2 * half;
    int bsel = (lm < BATCH) ? lm : 0;   // lanes N>=2 compute garbage cols (never stored)
    const float* arow = act + (size_t)bsel * (size_t)K + 2 * half;

    v8f c = {};
    for (int k0 = kbegin; k0 < kend; k0 += 16) {
        __builtin_prefetch(wrow + k0 + 128, 0, 3);   // global_prefetch ~512B ahead
        #pragma unroll
        for (int kk = 0; kk < 16; kk += 4) {
            v2f a, b;
            a.x = wrow[k0 + kk];
            a.y = wrow[k0 + kk + 1];
            b.x = arow[k0 + kk];
            b.y = arow[k0 + kk + 1];
            c = __builtin_amdgcn_wmma_f32_16x16x4_f32(false, a, false, b,
                                                      (short)0, c, false, false);
        }
    }

    if (lm < BATCH) {
        float* p = part + ((size_t)chunk * BATCH + lm) * (size_t)rows;
        #pragma unroll
        for (int v = 0; v < 8; ++v) {
            int row = row_base + v + 8 * half;   // C/D layout: M = v + 8*half, N = lm
            p[row] = c[v];
        }
    }
}

// Deterministic split-K reduction + epilogue.
// mode: 0 = +bias, 1 = +bias +resid, 2 = gelu
__global__ __launch_bounds__(256)
void gemv_reduce_kernel(const float* __restrict__ part,
                        const float* __restrict__ bias,
                        const float* __restrict__ resid,
                        float* __restrict__ out,
                        int rows, int mode) {
    int tid = blockIdx.x * blockDim.x + threadIdx.x;
    if (tid >= BATCH * rows) return;
    int b = tid / rows, r = tid % rows;
    float acc = 0.f;
    #pragma unroll
    for (int c = 0; c < NSPLIT; ++c)
        acc += part[((size_t)c * BATCH + b) * (size_t)rows + r];
    if (bias) acc += bias[r];
    if (mode == 1 && resid) acc += resid[(size_t)b * rows + r];
    if (mode == 2) acc = gelu_tanh(acc);
    out[(size_t)b * rows + r] = acc;
}

// ---------------------------------------------------------------------------
// RMSNorm: one block per batch element
// ---------------------------------------------------------------------------
__global__ __launch_bounds__(256)
void rmsnorm_kernel(const float* __restrict__ x, const float* __restrict__ w,
                    float* __restrict__ xn) {
    int b = blockIdx.x;
    __shared__ float red[256];
    float s = 0.f;
    for (int j = threadIdx.x; j < Hh; j += 256) {
        float v = x[b * Hh + j];
        s += v * v;
    }
    red[threadIdx.x] = s;
    __syncthreads();
    for (int o = 128; o > 0; o >>= 1) {
        if (threadIdx.x < o) red[threadIdx.x] += red[threadIdx.x + o];
        __syncthreads();
    }
    float scale = rsqrtf(red[0] * (1.0f / Hh) + EPSV);
    for (int j = threadIdx.x; j < Hh; j += 256)
        xn[b * Hh + j] = x[b * Hh + j] * scale * w[j];
}

// short 3-tap FIR on a single z channel j (cache=2)
__device__ __forceinline__ float sfir3(const float* __restrict__ z,
                                       const float* __restrict__ sfw,
                                       const float* __restrict__ sfb,
                                       const float* __restrict__ fst,
                                       int b, int j) {
    float u = z[b * H3 + j];
    const float* w  = sfw + (size_t)j * 3;
    const float* st = fst + ((size_t)b * H3 + j) * 2;
    return w[2] * u + st[0] * w[0] + st[1] * w[1] + sfb[j];
}

// ---------------------------------------------------------------------------
// HCS mix: 7-tap FIR, not flipped, b=D (ungated)
// ---------------------------------------------------------------------------
__global__ __launch_bounds__(256)
void mix_hcs_kernel(const float* __restrict__ z, const float* __restrict__ sfw,
                    const float* __restrict__ sfb, const float* __restrict__ fst,
                    const float* __restrict__ hw, const float* __restrict__ D,
                    const float* __restrict__ st, float* __restrict__ y) {
    int tid = blockIdx.x * blockDim.x + threadIdx.x;
    if (tid >= BATCH * Hh) return;
    int b = tid / Hh, hh = tid % Hh;
    int n = hh >> 6, cc = hh & 63, base = n * 192;
    float x2 = sfir3(z, sfw, sfb, fst, b, base + cc);
    float x1 = sfir3(z, sfw, sfb, fst, b, base + 64 + cc);
    float vv = sfir3(z, sfw, sfb, fst, b, base + 128 + cc);
    float u = x1 * vv;
    const float* w = hw + (size_t)hh * 7;
    const float* s = st + ((size_t)b * Hh + hh) * 6;
    float acc = w[6] * u + D[hh];
    #pragma unroll
    for (int t = 0; t < 6; ++t) acc += s[t] * w[t];
    y[(size_t)b * Hh + hh] = acc * x2;
}

// ---------------------------------------------------------------------------
// HCM mix: 128-tap FIR, flipped taps, gated (b = D*u)
// ---------------------------------------------------------------------------
__global__ __launch_bounds__(256)
void mix_hcm_kernel(const float* __restrict__ z, const float* __restrict__ sfw,
                    const float* __restrict__ sfb, const float* __restrict__ fst,
                    const float* __restrict__ hw, const float* __restrict__ D,
                    const float* __restrict__ st, float* __restrict__ y) {
    int tid = blockIdx.x * blockDim.x + threadIdx.x;
    if (tid >= BATCH * Hh) return;
    int b = tid / Hh, hh = tid % Hh;
    int n = hh >> 6, cc = hh & 63, base = n * 192;
    float x2 = sfir3(z, sfw, sfb, fst, b, base + cc);
    float x1 = sfir3(z, sfw, sfb, fst, b, base + 64 + cc);
    float vv = sfir3(z, sfw, sfb, fst, b, base + 128 + cc);
    float u = x1 * vv;
    const float* w = hw + (size_t)hh * 128;
    const float* s = st + ((size_t)b * Hh + hh) * 127;
    float acc = w[0] * u + D[hh] * u;     // flipped: w_eff[t] = w[127-t]
    for (int t = 0; t < 127; ++t) acc += s[t] * w[127 - t];
    y[(size_t)b * Hh + hh] = acc * x2;
}

// ---------------------------------------------------------------------------
// HCL mix: 16-pole IIR
// ---------------------------------------------------------------------------
__global__ __launch_bounds__(256)
void mix_hcl_kernel(const float* __restrict__ z, const float* __restrict__ sfw,
                    const float* __restrict__ sfb, const float* __restrict__ fst,
                    const float* __restrict__ lp, const float* __restrict__ rs,
                    const float* __restrict__ D, const float* __restrict__ iir,
                    float* __restrict__ y) {
    int tid = blockIdx.x * blockDim.x + threadIdx.x;
    if (tid >= BATCH * Hh) return;
    int b = tid / Hh, hh = tid % Hh;
    int n = hh >> 6, cc = hh & 63, base = n * 192;
    float x2 = sfir3(z, sfw, sfb, fst, b, base + cc);
    float x1 = sfir3(z, sfw, sfb, fst, b, base + 64 + cc);
    float vv = sfir3(z, sfw, sfb, fst, b, base + 128 + cc);
    float u = x1 * vv;
    const float* l = lp + (size_t)hh * 16;
    const float* r = rs + (size_t)hh * 16;
    const float* s = iir + ((size_t)b * Hh + hh) * 16;
    float acc = 0.f;
    #pragma unroll
    for (int p = 0; p < 16; ++p) acc += r[p] * (expf(l[p]) * s[p] + u);
    y[(size_t)b * Hh + hh] = x2 * (acc + D[hh] * u);
}

// ---------------------------------------------------------------------------
// RoPE on q,k from the projected z=[B][3H]; v copied.
// ---------------------------------------------------------------------------
__global__ __launch_bounds__(256)
void rope_kernel(const float* __restrict__ z,
                 const float* __restrict__ cosT, const float* __restrict__ sinT,
                 const int* __restrict__ posp,
                 float* __restrict__ qb, float* __restrict__ kb,
                 float* __restrict__ vb) {
    int tid = blockIdx.x * blockDim.x + threadIdx.x;
    if (tid >= BATCH * Hh) return;
    int b = tid / Hh, j = tid % Hh;
    int d = j & 63;
    int pos = *posp;
    float c = cosT[(size_t)pos * 32 + (d & 31)];
    float s = sinT[(size_t)pos * 32 + (d & 31)];
    float q  = z[(size_t)b * H3 + j];
    float qr = (d < 32) ? -z[(size_t)b * H3 + j + 32] : z[(size_t)b * H3 + j - 32];
    float k  = z[(size_t)b * H3 + Hh + j];
    float kr = (d < 32) ? -z[(size_t)b * H3 + Hh + j + 32] : z[(size_t)b * H3 + Hh + j - 32];
    qb[(size_t)b * Hh + j] = q * c + qr * s;
    kb[(size_t)b * Hh + j] = k * c + kr * s;
    vb[(size_t)b * Hh + j] = z[(size_t)b * H3 + 2 * Hh + j];
}

// ---------------------------------------------------------------------------
// Attention: one block per (batch, head). KV cache is never mutated; s==pos
// uses the freshly rotated k / new v. Masked s>pos underflows to exactly 0.
// ---------------------------------------------------------------------------
__global__ __launch_bounds__(256)
void attn_kernel(const float* __restrict__ qb, const float* __restrict__ kb,
                 const float* __restrict__ vb,
                 const float* __restrict__ kc, const float* __restrict__ vc,
                 const int* __restrict__ posp,
                 float* __restrict__ scores, float* __restrict__ y) {
    int b = blockIdx.x >> 4;
    int n = blockIdx.x & 15;
    int pos = *posp;
    const float* q = qb + ((size_t)b * NHEAD + n) * HDIM;
    float* sc = scores + ((size_t)b * NHEAD + n) * SEQ;

    __shared__ float red[256];
    __shared__ float ctxp[4][HDIM];

    float m = -1e30f;
    for (int s = threadIdx.x; s <= pos; s += 256) {
        const float* kr = (s == pos) ? (kb + ((size_t)b * NHEAD + n) * HDIM)
                                     : (kc + (((size_t)b * SEQ + s) * NHEAD + n) * HDIM);
        float acc = 0.f;
        #pragma unroll 8
        for (int d = 0; d < HDIM; ++d) acc += q[d] * kr[d];
        acc *= 0.125f;
        sc[s] = acc;
        m = fmaxf(m, acc);
    }
    red[threadIdx.x] = m; __syncthreads();
    for (int o = 128; o > 0; o >>= 1) {
        if (threadIdx.x < o) red[threadIdx.x] = fmaxf(red[threadIdx.x], red[threadIdx.x + o]);
        __syncthreads();
    }
    m = red[0]; __syncthreads();

    float ssum = 0.f;
    for (int s = threadIdx.x; s <= pos; s += 256) {
        float p = expf(sc[s] - m);
        sc[s] = p;
        ssum += p;
    }
    red[threadIdx.x] = ssum; __syncthreads();
    for (int o = 128; o > 0; o >>= 1) {
        if (threadIdx.x < o) red[threadIdx.x] += red[threadIdx.x + o];
        __syncthreads();
    }
    float denom = red[0];

    int d = threadIdx.x & 63, g = threadIdx.x >> 6;
    float acc = 0.f;
    for (int s = g; s <= pos; s += 4) {
        const float* vr = (s == pos) ? (vb + ((size_t)b * NHEAD + n) * HDIM)
                                     : (vc + (((size_t)b * SEQ + s) * NHEAD + n) * HDIM);
        acc += sc[s] * vr[d];
    }
    ctxp[g][d] = acc; __syncthreads();
    if (threadIdx.x < HDIM) {
        float c = (ctxp[0][threadIdx.x] + ctxp[1][threadIdx.x] +
                   ctxp[2][threadIdx.x] + ctxp[3][threadIdx.x]) / denom;
        y[(size_t)b * Hh + n * HDIM + threadIdx.x] = c;
    }
}

// ---------------------------------------------------------------------------
// host driver
// ---------------------------------------------------------------------------
static inline void gemv(const float* W, const float* bias, const float* act,
                        const float* resid, float* out, float* part,
                        int rows, int K, int mode, hipStream_t stream) {
    dim3 grid(rows >> 6, NSPLIT);   // 4 waves/block * 16 rows/wave = 64 rows/block
    gemv_wmma_partial<<<grid, 128, 0, stream>>>(W, act, part, rows, K);
    gemv_reduce_kernel<<<(BATCH * rows + 255) / 256, 256, 0, stream>>>(
        part, bias, resid, out, rows, mode);
}

extern "C" void kernel_launch(void* const* d_in, const int* in_sizes, int n_in,
                              void* d_out, int out_size, void* d_ws, size_t ws_size,
                              hipStream_t stream) {
    const float* x_in     = (const float*)d_in[0];
    const int*   posp     = (const int*)  d_in[1];
    const float* norm1_w  = (const float*)d_in[2];
    const float* norm2_w  = (const float*)d_in[3];
    const float* proj_W   = (const float*)d_in[4];
    const float* proj_b   = (const float*)d_in[5];
    const float* out_W    = (const float*)d_in[6];
    const float* out_b    = (const float*)d_in[7];
    const float* mlp_W1   = (const float*)d_in[8];
    const float* mlp_W2   = (const float*)d_in[9];
    const float* sf_w     = (const float*)d_in[10];
    const float* sf_b     = (const float*)d_in[11];
    const float* hcs_h    = (const float*)d_in[12];
    const float* hcs_D    = (const float*)d_in[13];
    const float* hcm_h    = (const float*)d_in[14];
    const float* hcm_D    = (const float*)d_in[15];
    const float* hcl_lp   = (const float*)d_in[16];
    const float* hcl_rs   = (const float*)d_in[17];
    const float* hcl_D    = (const float*)d_in[18];
    const float* fir_st   = (const float*)d_in[19];
    const float* hcs_st   = (const float*)d_in[20];
    const float* hcm_st   = (const float*)d_in[21];
    const float* iir_st   = (const float*)d_in[22];
    const float* k_cache  = (const float*)d_in[23];
    const float* v_cache  = (const float*)d_in[24];
    const float* rope_cos = (const float*)d_in[25];
    const float* rope_sin = (const float*)d_in[26];

    float* ws    = (float*)d_ws;
    float* x_cur = ws;                  // [B][H]
    float* xn    = ws + 2048;           // [B][H]
    float* z     = ws + 4096;           // [B][3H]
    float* y     = ws + 10240;          // [B][H]
    float* hbuf  = ws + 12288;          // [B][F]
    float* qb    = ws + 16384;          // [B][H]
    float* kb    = ws + 18432;
    float* vb    = ws + 20480;
    float* sc    = ws + 22528;          // [B][NH][S] = 65536 floats
    float* part  = ws + 88064;          // [NSPLIT][B][3072] max = 49152 floats

    // 0=HCS 1=HCM 2=HCL 3=ATT
    static const int TYPE[LAYERS] = {0,1,2,3,0,1,2,0,1,2,3,0,1,2,0,1,2,3,
                                     0,1,2,0,1,2,3,0,1,2,0,1,2,3};

    hipMemcpyAsync(x_cur, x_in, (size_t)BATCH * Hh * sizeof(float),
                   hipMemcpyDeviceToDevice, stream);

    int att = 0;
    for (int i = 0; i < LAYERS; ++i) {
        const float* fst = fir_st + (size_t)i * BATCH * H3 * 2;

        rmsnorm_kernel<<<BATCH, 256, 0, stream>>>(x_cur, norm1_w + (size_t)i * Hh, xn);
        gemv(proj_W + (size_t)i * H3 * Hh, proj_b + (size_t)i * H3,
             xn, nullptr, z, part, H3, Hh, 0, stream);

        if (TYPE[i] == 0) {
            mix_hcs_kernel<<<(BATCH * Hh + 255) / 256, 256, 0, stream>>>(
                z, sf_w + (size_t)i * H3 * 3, sf_b + (size_t)i * H3, fst,
                hcs_h + (size_t)i * Hh * 7, hcs_D + (size_t)i * Hh,
                hcs_st + (size_t)i * BATCH * Hh * 6, y);
        } else if (TYPE[i] == 1) {
            mix_hcm_kernel<<<(BATCH * Hh + 255) / 256, 256, 0, stream>>>(
                z, sf_w + (size_t)i * H3 * 3, sf_b + (size_t)i * H3, fst,
                hcm_h + (size_t)i * Hh * 128, hcm_D + (size_t)i * Hh,
                hcm_st + (size_t)i * BATCH * Hh * 127, y);
        } else if (TYPE[i] == 2) {
            mix_hcl_kernel<<<(BATCH * Hh + 255) / 256, 256, 0, stream>>>(
                z, sf_w + (size_t)i * H3 * 3, sf_b + (size_t)i * H3, fst,
                hcl_lp + (size_t)i * Hh * 16, hcl_rs + (size_t)i * Hh * 16,
                hcl_D + (size_t)i * Hh,
                iir_st + (size_t)i * BATCH * Hh * 16, y);
        } else {
            rope_kernel<<<(BATCH * Hh + 255) / 256, 256, 0, stream>>>(
                z, rope_cos, rope_sin, posp, qb, kb, vb);
            attn_kernel<<<BATCH * NHEAD, 256, 0, stream>>>(
                qb, kb, vb,
                k_cache + (size_t)att * BATCH * SEQ * NHEAD * HDIM,
                v_cache + (size_t)att * BATCH * SEQ * NHEAD * HDIM,
                posp, sc, y);
            ++att;
        }

        gemv(out_W + (size_t)i * Hh * Hh, out_b + (size_t)i * Hh,
             y, x_cur, x_cur, part, Hh, Hh, 1, stream);

        rmsnorm_kernel<<<BATCH, 256, 0, stream>>>(x_cur, norm2_w + (size_t)i * Hh, xn);
        gemv(mlp_W1 + (size_t)i * Ff * Hh, nullptr, xn, nullptr, hbuf, part,
             Ff, Hh, 2, stream);
        gemv(mlp_W2 + (size_t)i * Hh * Ff, nullptr, hbuf, x_cur, x_cur, part,
             Hh, Ff, 1, stream);
    }

    hipMemcpyAsync(d_out, x_cur, (size_t)BATCH * Hh * sizeof(float),
                   hipMemcpyDeviceToDevice, stream);
}